// MemorySeCo_35759897707142
// MI455X (gfx1250) — compile-verified
//
#include <hip/hip_runtime.h>

typedef __attribute__((ext_vector_type(2))) float v2f;
typedef __attribute__((ext_vector_type(8))) float v8f;

#define BATCH     512
#define FDIM      128
#define QUEUE     65536
#define NEG_ROWS  8192            // B*N = 512*16 in-batch negatives
#define NTOT      (NEG_ROWS + QUEUE)   // 73728
#define OUT_COLS  (1 + NTOT)           // 73729
#define INV_TEMP  10.0f

// ---------------------------------------------------------------------------
// GEMM: l_neg = q @ [neg_set; memory]^T, scaled by 1/TEMP, written to BOTH
// row-halves of out (cols 1..73728). One wave per 16(M) x 64(N) strip.
// Uses V_WMMA_F32_16X16X4_F32, K=128 in 32 steps of 4.
// ---------------------------------------------------------------------------
__global__ __launch_bounds__(256) void seco_gemm_kernel(
    const float* __restrict__ q,        // [512,128]
    const float* __restrict__ neg_set,  // [8192,128] (flattened [B,N,D])
    const float* __restrict__ memory,   // [65536,128]
    float* __restrict__ out)            // [1024,73729]
{
    const int wave = blockIdx.x * (blockDim.x >> 5) + (threadIdx.x >> 5);
    const int lane = threadIdx.x & 31;

    const int mTiles = BATCH / 16;        // 32
    const int tm = wave & (mTiles - 1);   // M tile index
    const int tg = wave >> 5;             // 64-column group index
    if (tg >= NTOT / 64) return;

    const int row   = lane & 15;          // M-row (A) / N-col (B,C)
    const int khalf = (lane >> 4) << 1;   // 0 or 2: which K pair this lane holds

    // ---- Preload A tile (16x128 of q) into registers: one float2 per K-step.
    const float* qrow = q + (tm * 16 + row) * FDIM;
    v2f areg[32];
#pragma unroll
    for (int s = 0; s < 32; ++s) {
        const float2 t = *reinterpret_cast<const float2*>(qrow + 4 * s + khalf);
        areg[s].x = t.x;
        areg[s].y = t.y;
    }

    // ---- B base: virtual concat [neg_set; memory] splits at row 8192 (mult of 64).
    const int col0 = tg * 64;
    const float* nbase = (col0 < NEG_ROWS) ? neg_set : memory;
    const int    nrow0 = (col0 < NEG_ROWS) ? col0 : (col0 - NEG_ROWS);

    const float* brow0 = nbase + (size_t)(nrow0 +  0 + row) * FDIM;
    const float* brow1 = nbase + (size_t)(nrow0 + 16 + row) * FDIM;
    const float* brow2 = nbase + (size_t)(nrow0 + 32 + row) * FDIM;
    const float* brow3 = nbase + (size_t)(nrow0 + 48 + row) * FDIM;

    v8f acc[4] = {};

#pragma unroll
    for (int s = 0; s < 32; ++s) {
        const int ko = 4 * s + khalf;
        float2 t;
        v2f b;
        t = *reinterpret_cast<const float2*>(brow0 + ko);
        b.x = t.x; b.y = t.y;
        acc[0] = __builtin_amdgcn_wmma_f32_16x16x4_f32(
            false, areg[s], false, b, (short)0, acc[0], false, false);
        t = *reinterpret_cast<const float2*>(brow1 + ko);
        b.x = t.x; b.y = t.y;
        acc[1] = __builtin_amdgcn_wmma_f32_16x16x4_f32(
            false, areg[s], false, b, (short)0, acc[1], false, false);
        t = *reinterpret_cast<const float2*>(brow2 + ko);
        b.x = t.x; b.y = t.y;
        acc[2] = __builtin_amdgcn_wmma_f32_16x16x4_f32(
            false, areg[s], false, b, (short)0, acc[2], false, false);
        t = *reinterpret_cast<const float2*>(brow3 + ko);
        b.x = t.x; b.y = t.y;
        acc[3] = __builtin_amdgcn_wmma_f32_16x16x4_f32(
            false, areg[s], false, b, (short)0, acc[3], false, false);
    }

    // ---- Store. C/D layout: VGPR v -> M = v + 8*(lane>=16), N = lane&15.
    // Write each value to both row-halves of out (the tile(l_neg,(2,1))).
    const int mBase = tm * 16 + ((lane >> 4) << 3);
#pragma unroll
    for (int sub = 0; sub < 4; ++sub) {
        const size_t c = 1u + (size_t)col0 + sub * 16 + row;
#pragma unroll
        for (int v = 0; v < 8; ++v) {
            const float val = acc[sub][v] * INV_TEMP;
            float* p = out + (size_t)(mBase + v) * OUT_COLS + c;
            p[0] = val;
            p[(size_t)BATCH * OUT_COLS] = val;
        }
    }
}

// ---------------------------------------------------------------------------
// Positive logits: out[b][0]   = dot(q[b],k[b]) / TEMP
//                  out[b+B][0] = mean_p dot(q[b],pos_set[b,p]) / TEMP
// One wave per batch row; float4 loads + shfl_xor reduction (wave32).
// ---------------------------------------------------------------------------
__global__ __launch_bounds__(256) void seco_pos_kernel(
    const float* __restrict__ q,
    const float* __restrict__ k,
    const float* __restrict__ pos_set,  // [512,4,128]
    float* __restrict__ out)
{
    const int b    = blockIdx.x * (blockDim.x >> 5) + (threadIdx.x >> 5);
    const int lane = threadIdx.x & 31;
    if (b >= BATCH) return;

    const float4 qv = *reinterpret_cast<const float4*>(q + b * FDIM + lane * 4);
    const float4 kv = *reinterpret_cast<const float4*>(k + b * FDIM + lane * 4);
    float dk = qv.x * kv.x + qv.y * kv.y + qv.z * kv.z + qv.w * kv.w;

    float dp = 0.0f;
#pragma unroll
    for (int p = 0; p < 4; ++p) {
        const float4 pv = *reinterpret_cast<const float4*>(
            pos_set + ((size_t)b * 4 + p) * FDIM + lane * 4);
        dp += qv.x * pv.x + qv.y * pv.y + qv.z * pv.z + qv.w * pv.w;
    }

#pragma unroll
    for (int off = 16; off > 0; off >>= 1) {
        dk += __shfl_xor(dk, off, 32);
        dp += __shfl_xor(dp, off, 32);
    }

    if (lane == 0) {
        out[(size_t)b * OUT_COLS]           = dk * INV_TEMP;
        out[(size_t)(b + BATCH) * OUT_COLS] = (dp * 0.25f) * INV_TEMP;
    }
}

// ---------------------------------------------------------------------------
// new_memory = memory with rows (index..index+511) mod QUEUE <- k_all.
// Single fused pass, float4 per thread.
// ---------------------------------------------------------------------------
__global__ __launch_bounds__(256) void seco_queue_kernel(
    const float* __restrict__ memory,
    const float* __restrict__ k_all,
    const int* __restrict__ index_p,
    float* __restrict__ new_memory)
{
    const int idx = *index_p;
    const size_t tid = (size_t)blockIdx.x * blockDim.x + threadIdx.x; // QUEUE*32 float4s
    const int r  = (int)(tid >> 5);
    const int c4 = (int)(tid & 31) * 4;
    const int j  = (r - idx) & (QUEUE - 1);   // (r - index) mod 65536, power-of-2 wrap

    const float* src = (j < BATCH)
        ? (k_all + (size_t)j * FDIM + c4)
        : (memory + (size_t)r * FDIM + c4);
    *reinterpret_cast<float4*>(new_memory + (size_t)r * FDIM + c4) =
        *reinterpret_cast<const float4*>(src);
}

// ---------------------------------------------------------------------------
extern "C" void kernel_launch(void* const* d_in, const int* in_sizes, int n_in,
                              void* d_out, int out_size, void* d_ws, size_t ws_size,
                              hipStream_t stream) {
    const float* q       = (const float*)d_in[0];
    const float* k       = (const float*)d_in[1];
    const float* pos_set = (const float*)d_in[2];
    const float* neg_set = (const float*)d_in[3];
    const float* k_all   = (const float*)d_in[4];
    const float* memory  = (const float*)d_in[5];
    const int*   index   = (const int*)d_in[6];

    float* out        = (float*)d_out;                          // [1024, 73729]
    float* new_memory = out + (size_t)(2 * BATCH) * OUT_COLS;   // [65536, 128]

    // GEMM: 32 M-tiles * 1152 N-groups = 36864 waves; 8 waves/block -> 4608 blocks.
    seco_gemm_kernel<<<4608, 256, 0, stream>>>(q, neg_set, memory, out);

    // Positive logits: 512 waves -> 64 blocks of 8 waves.
    seco_pos_kernel<<<64, 256, 0, stream>>>(q, k, pos_set, out);

    // Queue copy/update: 65536*32 float4 threads / 256 = 8192 blocks.
    seco_queue_kernel<<<8192, 256, 0, stream>>>(memory, k_all, index, new_memory);
}